// QuantumClassifier_16234976378954
// MI455X (gfx1250) — compile-verified
//
#include <hip/hip_runtime.h>
#include <math.h>

// CDNA5 wave32 WMMA vectors
typedef __attribute__((ext_vector_type(2))) float v2f;
typedef __attribute__((ext_vector_type(8))) float v8f;

#define NQ   13
#define NST  8192      // 2^13 amplitudes
#define BLK  256       // 8 wave32

// ---------------- compile-time GF(2) linear maps for the CNOT cascades ----
// CNOT(c,t): new[i] = old[sigma(i)], sigma flips target bit if control set.
// After cascade q=0..12 (applied in order), state' = state o T where
// T(i) = sigma_first(...sigma_last(i)): apply last CNOT's sigma first.
struct LinMap { int col[NQ]; };

constexpr int csigma(int i, int c, int t) {
    return i ^ (((i >> (12 - c)) & 1) << (12 - t));
}
constexpr int cTlayer(int i, int r) {
    int j = i;
    for (int q = NQ - 1; q >= 0; --q) j = csigma(j, q, (q + r) % NQ);
    return j;
}
constexpr LinMap mkT(int r) {
    LinMap m{};
    for (int b = 0; b < NQ; ++b) m.col[b] = cTlayer(1 << b, r);
    return m;
}
constexpr LinMap mkPF() {   // PF = T0 o T1  (layer0 r=1, layer1 r=2)
    LinMap m{};
    for (int b = 0; b < NQ; ++b) m.col[b] = cTlayer(cTlayer(1 << b, 2), 1);
    return m;
}
static constexpr LinMap kT0 = mkT(1);
static constexpr LinMap kPF = mkPF();

__device__ __forceinline__ int applyP(int i, const LinMap& M) {
    int j = 0;
#pragma unroll
    for (int b = 0; b < NQ; ++b) j ^= (-((i >> b) & 1)) & M.col[b];
    return j;
}

// complex 2x2 Rot gate  U = RZ(om) RY(th) RZ(phi)
struct U2 { float r00, i00, r01, i01, r10, i10, r11, i11; };
__device__ __forceinline__ U2 mkU(const float* wg) {
    float phi = wg[0], th = wg[1], om = wg[2];
    float ch = cosf(0.5f * th), sh = sinf(0.5f * th);
    float sp = 0.5f * (phi + om), sm = 0.5f * (phi - om);
    float csp = cosf(sp), ssp = sinf(sp);
    float csm = cosf(sm), ssm = sinf(sm);
    U2 u;
    u.r00 =  csp * ch; u.i00 = -ssp * ch;
    u.r01 = -csm * sh; u.i01 = -ssm * sh;
    u.r10 =  csm * sh; u.i10 = -ssm * sh;
    u.r11 =  csp * ch; u.i11 =  ssp * ch;
    return u;
}

// One workgroup simulates one batch sample entirely in LDS (64 KB).
__global__ __launch_bounds__(BLK) void qsim_kernel(const float* __restrict__ x,
                                                   const float* __restrict__ w,
                                                   float* __restrict__ out) {
    __shared__ float sre[NST];
    __shared__ float sim[NST];

    const int tid  = threadIdx.x;
    const int lane = tid & 31;
    const int wave = tid >> 5;
    const int b    = blockIdx.x;
    const bool hiK = (lane & 16) != 0;

    const float* xb = x + (size_t)b * NST;

    // ---- AmplitudeEmbedding: load x, compute norm, psi = x/norm (+0i) ----
    float ss = 0.f;
    for (int k = 0; k < NST / BLK; ++k) {
        int i = tid + k * BLK;
        float v = xb[i];
        sre[i] = v;
        ss += v * v;
    }
    for (int m = 16; m; m >>= 1) ss += __shfl_xor(ss, m, 32);
    __syncthreads();
    if (lane == 0) sim[wave] = ss;
    __syncthreads();
    float tot = 0.f;
    for (int k = 0; k < 8; ++k) tot += sim[k];
    float inv = 1.0f / sqrtf(tot);
    __syncthreads();
    for (int k = 0; k < NST / BLK; ++k) {
        int i = tid + k * BLK;
        sre[i] *= inv;
        sim[i] = 0.f;
    }
    __syncthreads();

    const v8f zeroC = {};

#pragma unroll
    for (int l = 0; l < 2; ++l) {
        const bool ident = (l == 0);   // layer-0 gates see identity layout;
                                       // layer-1 gates see perm T0 (layer-0 CNOTs)

        // -------- fused 2-qubit gates on (2g, 2g+1): two chained WMMAs ----
#pragma unroll
        for (int g = 0; g < 6; ++g) {
            const int q = 2 * g;
            U2 Ua = mkU(w + (l * NQ + q) * 3);       // higher bit of quad
            U2 Ub = mkU(w + (l * NQ + q + 1) * 3);   // lower bit of quad

            // A(16x4) halves of the real 8x8 of Ua (x) Ub, rows = comps 0..7 x2.
            // Lane L row L%8; lane<16 -> K 0,1 ; lane>=16 -> K 2,3 (per chunk).
            const int  rr = lane & 7;
            const bool ri = (rr & 1) != 0;     // re/im output row
            const bool rA = (rr & 4) != 0;     // Ua row
            const bool rB = (rr & 2) != 0;     // Ub row
            // Ua[rA][0], Ua[rA][1]
            float qar = rA ? Ua.r10 : Ua.r00, qai = rA ? Ua.i10 : Ua.i00;
            float qbr = rA ? Ua.r11 : Ua.r01, qbi = rA ? Ua.i11 : Ua.i01;
            // Ub[rB][hiK]
            float wr = rB ? (hiK ? Ub.r11 : Ub.r10) : (hiK ? Ub.r01 : Ub.r00);
            float wi = rB ? (hiK ? Ub.i11 : Ub.i10) : (hiK ? Ub.i01 : Ub.i00);
            // V1 = Ua[rA][0]*Ub[rB][hiK]  (K-chunk 0);  V2 = Ua[rA][1]*... (chunk 1)
            float V1r = qar * wr - qai * wi, V1i = qar * wi + qai * wr;
            float V2r = qbr * wr - qbi * wi, V2i = qbr * wi + qbi * wr;
            v2f A1, A2;
            A1.x = ri ? V1i : V1r;  A1.y = ri ? V1r : -V1i;
            A2.x = ri ? V2i : V2r;  A2.y = ri ? V2r : -V2i;

            const int m1 = 1 << (12 - q);      // logical stride of qubit q
            const int m2 = m1 >> 1;            // logical stride of qubit q+1
            const int offs1 = ident ? m1 : kT0.col[12 - q];
            const int offs2 = ident ? m2 : kT0.col[11 - q];

            for (int it = 0; it < (NST / 4) / 128; ++it) {   // 16 iters
                int p    = it * 128 + wave * 16 + (lane & 15);
                int low  = p & (m2 - 1);
                int idx0 = ((p ^ low) << 2) | low;           // insert 2 zero bits
                int base = ident ? idx0 : applyP(idx0, kT0);
                int pa = base ^ (hiK ? offs2 : 0);           // amp0/amp1
                int pb = pa ^ offs1;                          // amp2/amp3
                v2f B1; B1.x = sre[pa]; B1.y = sim[pa];
                v2f B2; B2.x = sre[pb]; B2.y = sim[pb];
                v8f D = __builtin_amdgcn_wmma_f32_16x16x4_f32(
                    false, A1, false, B1, (short)0, zeroC, false, false);
                D = __builtin_amdgcn_wmma_f32_16x16x4_f32(
                    false, A2, false, B2, (short)0, D, false, false);
                sre[pa] = hiK ? D[2] : D[0];
                sim[pa] = hiK ? D[3] : D[1];
                sre[pb] = hiK ? D[6] : D[4];
                sim[pb] = hiK ? D[7] : D[5];
            }
            __syncthreads();
        }

        // -------- remaining single gate on qubit 12 (logical stride 1) ----
        {
            U2 U = mkU(w + (l * NQ + 12) * 3);
            int  j   = lane & 3;
            bool odd = (j & 1) != 0;
            bool up  = (j & 2) != 0;
            float xr = up ? U.r10 : U.r00, xi = up ? U.i10 : U.i00;
            float yr = up ? U.r11 : U.r01, yi = up ? U.i11 : U.i01;
            float a0 = odd ? xi :  xr;
            float a1 = odd ? xr : -xi;
            float a2 = odd ? yi :  yr;
            float a3 = odd ? yr : -yi;
            v2f A;
            A.x = hiK ? a2 : a0;
            A.y = hiK ? a3 : a1;

            const int offs = ident ? 1 : kT0.col[0];
            for (int it = 0; it < (NST / 2) / 128; ++it) {   // 32 iters
                int p    = it * 128 + wave * 16 + (lane & 15);
                int idx0 = p << 1;
                int base = ident ? idx0 : applyP(idx0, kT0);
                int pa   = base ^ (hiK ? offs : 0);
                v2f B; B.x = sre[pa]; B.y = sim[pa];
                v8f D = __builtin_amdgcn_wmma_f32_16x16x4_f32(
                    false, A, false, B, (short)0, zeroC, false, false);
                sre[pa] = hiK ? D[2] : D[0];
                sim[pa] = hiK ? D[3] : D[1];
            }
            __syncthreads();
        }
        // layer's CNOT cascade: absorbed into the compile-time index maps.
    }

    // ---- expval(PauliZ(qubit 6)): gather through PF, sign from logical bit 6
    float es = 0.f;
    for (int k = 0; k < NST / BLK; ++k) {
        int i = tid + k * BLK;
        int j = applyP(i, kPF);
        float pr = sre[j] * sre[j] + sim[j] * sim[j];
        es += (i & 64) ? -pr : pr;
    }
    for (int m = 16; m; m >>= 1) es += __shfl_xor(es, m, 32);
    __syncthreads();
    if (lane == 0) sre[wave] = es;
    __syncthreads();
    if (tid == 0) {
        float acc = 0.f;
        for (int k = 0; k < 8; ++k) acc += sre[k];
        out[b] = acc;
    }
}

extern "C" void kernel_launch(void* const* d_in, const int* in_sizes, int n_in,
                              void* d_out, int out_size, void* d_ws, size_t ws_size,
                              hipStream_t stream) {
    (void)n_in; (void)d_ws; (void)ws_size;
    const float* x = (const float*)d_in[0];      // (B, 8192) f32
    const float* w = (const float*)d_in[1];      // (2, 13, 3) f32
    float* out = (float*)d_out;                  // (B,) f32
    int B = in_sizes[0] / NST;
    if (B <= 0) B = out_size;
    qsim_kernel<<<dim3(B), dim3(BLK), 0, stream>>>(x, w, out);
}